// HeteroGNN_22033182228530
// MI455X (gfx1250) — compile-verified
//
#include <hip/hip_runtime.h>

typedef __attribute__((ext_vector_type(2))) float v2f;
typedef __attribute__((ext_vector_type(8))) float v8f;

#define DFEAT 64

// D = A(16x4) * B(4x16) + C, fp32 WMMA (full precision, runs on matrix pipe)
__device__ __forceinline__ v8f wmma16x16x4(v2f a, v2f b, v8f c) {
#if defined(__HIP_DEVICE_COMPILE__)
  // 8-arg pattern: (neg_a, A, neg_b, B, c_mod, C, reuse_a, reuse_b)
  return __builtin_amdgcn_wmma_f32_16x16x4_f32(false, a, false, b, (short)0, c,
                                               false, false);
#else
  (void)a; (void)b;
  return c;
#endif
}

__global__ void zero_f32(float* __restrict__ p, long n) {
  long i = (long)blockIdx.x * blockDim.x + threadIdx.x;
  long stride = (long)gridDim.x * blockDim.x;
  for (; i < n; i += stride) p[i] = 0.0f;
}

// One edge handled by 16 threads; each thread moves a float4 chunk (16B).
// Segment-sum via fp32 atomics; count accumulated once per edge.
__global__ void scatter_accum(const float* __restrict__ xsrc,
                              const int* __restrict__ src,
                              const int* __restrict__ dst,
                              float* __restrict__ sum,
                              float* __restrict__ cnt,
                              int nE, int addcnt) {
  long gid = (long)blockIdx.x * blockDim.x + threadIdx.x;
  int e = (int)(gid >> 4);
  if (e >= nE) return;
  int c = (int)(gid & 15);
  int s = src[e];
  int d = dst[e];
  const float4 v = *(const float4*)(xsrc + (long)s * DFEAT + c * 4);
  float* dp = sum + (long)d * DFEAT + c * 4;
  atomicAdd(dp + 0, v.x);
  atomicAdd(dp + 1, v.y);
  atomicAdd(dp + 2, v.z);
  atomicAdd(dp + 3, v.w);
  if (addcnt && c == 0) atomicAdd(cnt + d, 1.0f);
}

// h[v] = act( (sum[v]/max(cnt[v],1)) @ Wl + xroot[v] @ Wr + b )
// One wave per 16-row tile; 4 N-tiles of 16 cols; K=64 in steps of 4.
// Weights staged into LDS in K-paired layout so each WMMA B operand is a
// single aligned ds_load_b64 into an adjacent (even) VGPR pair:
//   sW[((k>>1)*64 + col)*2 + (k&1)] = W[k*64 + col]
// NOTE: outh may alias aggsum (in-place): each wave reads all its rows
// across the K loop before storing, so no restrict on those pointers.
__global__ __launch_bounds__(256) void sage_wmma(
    const float* aggsum, const float* __restrict__ cnt, const float* xroot,
    const float* __restrict__ Wl, const float* __restrict__ Wr,
    const float* __restrict__ bias, float* outh, int ntiles, int do_relu) {
  __shared__ float sWl[DFEAT * DFEAT];
  __shared__ float sWr[DFEAT * DFEAT];
  for (int i = threadIdx.x; i < DFEAT * DFEAT; i += blockDim.x) {
    int k = i >> 6;        // row (K index) of W
    int col = i & 63;      // output column
    int pidx = (((k >> 1) * DFEAT + col) << 1) | (k & 1);
    sWl[pidx] = Wl[i];
    sWr[pidx] = Wr[i];
  }
  __syncthreads();

  int wave = threadIdx.x >> 5;
  int lane = threadIdx.x & 31;
  int tile = blockIdx.x * 8 + wave;
  if (tile >= ntiles) return;

  int m0 = tile * 16;
  int mrow = m0 + (lane & 15);  // A: lanes 0-15/16-31 both cover M=0..15
  int khalf = lane >> 4;        // 0 -> K {k,k+1}; 1 -> K {k+2,k+3}
  int ncol = lane & 15;         // B/C: N index within the 16-col tile
  float rcp = 1.0f / fmaxf(cnt[mrow], 1.0f);

  v8f acc[4] = {};

  const float* aggp = aggsum + (long)mrow * DFEAT;
  const float* xp = xroot + (long)mrow * DFEAT;

#pragma unroll
  for (int k0 = 0; k0 < DFEAT; k0 += 4) {
    int ka = k0 + 2 * khalf;  // even
    v2f Aagg;
    Aagg.x = aggp[ka] * rcp;
    Aagg.y = aggp[ka + 1] * rcp;
    v2f Ax;
    Ax.x = xp[ka];
    Ax.y = xp[ka + 1];
    int pbase = (ka >> 1) * (DFEAT * 2);  // paired-row base in LDS
#pragma unroll
    for (int nt = 0; nt < 4; ++nt) {
      int col = nt * 16 + ncol;
      int pidx = pbase + col * 2;  // 8B aligned, contiguous pair
      v2f Bl;
      Bl.x = sWl[pidx];
      Bl.y = sWl[pidx + 1];
      v2f Br;
      Br.x = sWr[pidx];
      Br.y = sWr[pidx + 1];
      acc[nt] = wmma16x16x4(Aagg, Bl, acc[nt]);
      acc[nt] = wmma16x16x4(Ax, Br, acc[nt]);
    }
  }

  // C/D layout: lanes 0-15 hold M=r, lanes 16-31 hold M=r+8 (VGPR r=0..7)
  int rh = lane >> 4;
#pragma unroll
  for (int nt = 0; nt < 4; ++nt) {
    int col = nt * 16 + ncol;
    float bb = bias[col];
#pragma unroll
    for (int r = 0; r < 8; ++r) {
      float v = acc[nt][r] + bb;
      if (do_relu) v = fmaxf(v, 0.0f);
      outh[(long)(m0 + rh * 8 + r) * DFEAT + col] = v;
    }
  }
}

__global__ void final_linear(const float* __restrict__ h,
                             const float* __restrict__ Wlin,
                             const float* __restrict__ blin,
                             float* __restrict__ out, int N) {
  int i = blockIdx.x * blockDim.x + threadIdx.x;
  if (i >= N) return;
  float a0 = blin[0], a1 = blin[1];
  const float* hp = h + (long)i * DFEAT;
#pragma unroll 8
  for (int j = 0; j < DFEAT; ++j) {
    float v = hp[j];
    a0 += v * Wlin[2 * j];
    a1 += v * Wlin[2 * j + 1];
  }
  out[2 * i] = a0;
  out[2 * i + 1] = a1;
}

extern "C" void kernel_launch(void* const* d_in, const int* in_sizes, int n_in,
                              void* d_out, int out_size, void* d_ws,
                              size_t ws_size, hipStream_t stream) {
  const float* x_c = (const float*)d_in[0];
  const float* x_m = (const float*)d_in[1];
  const int* cm_src = (const int*)d_in[2];
  const int* cm_dst = (const int*)d_in[3];
  const int* mc_src = (const int*)d_in[4];
  const int* mc_dst = (const int*)d_in[5];
  const float* Wl1_cm = (const float*)d_in[6];
  const float* Wr1_cm = (const float*)d_in[7];
  const float* b1_cm = (const float*)d_in[8];
  const float* Wl1_mc = (const float*)d_in[9];
  const float* Wr1_mc = (const float*)d_in[10];
  const float* b1_mc = (const float*)d_in[11];
  // d_in[12..14] = Wl2_cm/Wr2_cm/b2_cm feed h2_m, which is dead in the ref.
  const float* Wl2_mc = (const float*)d_in[15];
  const float* Wr2_mc = (const float*)d_in[16];
  const float* b2_mc = (const float*)d_in[17];
  const float* W_lin = (const float*)d_in[18];
  const float* b_lin = (const float*)d_in[19];
  float* out = (float*)d_out;

  const int NC = in_sizes[0] / DFEAT;  // 200000 (multiple of 16)
  const int NM = in_sizes[1] / DFEAT;  // 50000  (multiple of 16)
  const int E = in_sizes[2];           // 1000000

  // Workspace layout (fp32): ~116 MB total
  float* bufC = (float*)d_ws;                   // NC*64: sum_c -> h_c (in place)
  float* bufM = bufC + (size_t)NC * DFEAT;      // NM*64: sum_m -> h_m (in place)
  float* bufC2 = bufM + (size_t)NM * DFEAT;     // NC*64: sum2_c -> h2_c (in place)
  float* cntC = bufC2 + (size_t)NC * DFEAT;     // NC counts (mc edges, reused L2)
  float* cntM = cntC + NC;                      // NM counts (cm edges)
  long total = (long)NC * DFEAT + (long)NM * DFEAT + (long)NC * DFEAT + NC + NM;

  zero_f32<<<2048, 256, 0, stream>>>(bufC, total);

  int sblocks = (int)(((long)E * 16 + 255) / 256);
  // Layer 1 aggregations
  scatter_accum<<<sblocks, 256, 0, stream>>>(x_c, cm_src, cm_dst, bufM, cntM, E, 1);
  scatter_accum<<<sblocks, 256, 0, stream>>>(x_m, mc_src, mc_dst, bufC, cntC, E, 1);

  int tM = NM / 16, tC = NC / 16;
  // Layer 1 dense transforms (+ReLU), in place over the sum buffers
  sage_wmma<<<(tM + 7) / 8, 256, 0, stream>>>(bufM, cntM, x_m, Wl1_cm, Wr1_cm,
                                              b1_cm, bufM, tM, 1);
  sage_wmma<<<(tC + 7) / 8, 256, 0, stream>>>(bufC, cntC, x_c, Wl1_mc, Wr1_mc,
                                              b1_mc, bufC, tC, 1);

  // Layer 2: only the mc direction matters (h2_m is unused). Counts for mc
  // edges are identical to layer 1's cntC, so skip count accumulation.
  scatter_accum<<<sblocks, 256, 0, stream>>>(bufM, mc_src, mc_dst, bufC2, cntC, E, 0);
  sage_wmma<<<(tC + 7) / 8, 256, 0, stream>>>(bufC2, cntC, bufC, Wl2_mc, Wr2_mc,
                                              b2_mc, bufC2, tC, 0);

  // Final 64 -> 2 linear on customer nodes
  final_linear<<<(NC + 255) / 256, 256, 0, stream>>>(bufC2, W_lin, b_lin, out, NC);
}